// GaussianBlur_15358803050751
// MI455X (gfx1250) — compile-verified
//
#include <hip/hip_runtime.h>
#include <hip/hip_bf16.h>

// Depthwise 4x4 binomial FIR (upfirdn2d pad=(2,1)) on (8,128,256,256) fp32.
// Bandwidth-bound: ~537MB HBM traffic -> ~23us floor at 23.3 TB/s; 2.1 GFLOP is noise.
// CDNA5 path: async global->LDS b128 staging (ASYNCcnt, 512B/wave-instr),
// s_wait_asynccnt, wave32 blocks, non-temporal 128-bit output stores.

#define TILE_W        64
#define TILE_H        16
#define CHUNKS_PER_ROW 18   // 72 floats per LDS row: [x0-4, x0+68), 16B-aligned chunks
#define IN_H          19    // TILE_H + 3 (halo: 2 top, 1 bottom)
#define LDS_STRIDE    76    // floats; multiple of 4 keeps every chunk 16B-aligned
#define NTHREADS      256   // 8 wave32s

typedef __attribute__((ext_vector_type(4))) float v4f;
typedef __attribute__((ext_vector_type(4))) int   v4i;

#if defined(__AMDGCN__) && __has_builtin(__builtin_amdgcn_global_load_async_to_lds_b128)
#define ASYNC_B128 1
#else
#define ASYNC_B128 0
#endif
#if defined(__AMDGCN__) && __has_builtin(__builtin_amdgcn_global_load_async_to_lds_b32)
#define ASYNC_B32 1
#else
#define ASYNC_B32 0
#endif

typedef __attribute__((address_space(1))) int g_as_i32;   // global scalar int*
typedef __attribute__((address_space(3))) int l_as_i32;   // LDS scalar int*
typedef __attribute__((address_space(1))) v4i g_as_i128;  // global int4*
typedef __attribute__((address_space(3))) v4i l_as_i128;  // LDS int4*

__global__ __launch_bounds__(NTHREADS)
void blur4x4_kernel(const float* __restrict__ in, const float* __restrict__ kern,
                    float* __restrict__ out, int H, int W) {
    __shared__ float tile[IN_H * LDS_STRIDE];

    const int  tid = threadIdx.x;
    const int  x0  = blockIdx.x * TILE_W;
    const int  y0  = blockIdx.y * TILE_H;
    const long plane = blockIdx.z;

    const float* __restrict__ gplane = in  + plane * (long)H * (long)W;
    float*       __restrict__ oplane = out + plane * (long)H * (long)W;

    // ---- Stage input tile (zero-padded halo) into LDS, 16B chunks ----
    // Chunk starts are =0 (mod 4) and W,H are multiples of 4, so every chunk is
    // fully in-bounds (async b128 DMA) or fully OOB (16B ds_store of zeros).
    const int gx_base = x0 - 4;   // 16B-aligned
    const int gy_base = y0 - 2;
    for (int e = tid; e < IN_H * CHUNKS_PER_ROW; e += NTHREADS) {
        const int r  = e / CHUNKS_PER_ROW;
        const int cc = e - r * CHUNKS_PER_ROW;
        const int gy = gy_base + r;
        const int gx = gx_base + 4 * cc;
        float* lptr = &tile[r * LDS_STRIDE + 4 * cc];
        if (gy >= 0 && gy < H && gx >= 0 && gx < W) {
            float* gptr = const_cast<float*>(gplane + (long)gy * W + gx);
#if ASYNC_B128
            __builtin_amdgcn_global_load_async_to_lds_b128(
                (g_as_i128*)gptr, (l_as_i128*)lptr, 0 /*imm offset*/, 0 /*cpol*/);
#elif ASYNC_B32
#pragma unroll
            for (int q = 0; q < 4; ++q)
                __builtin_amdgcn_global_load_async_to_lds_b32(
                    (g_as_i32*)(gptr + q), (l_as_i32*)(lptr + q), 0, 0);
#else
            *(v4f*)lptr = *(const v4f*)gptr;
#endif
        } else {
            v4f z = {0.f, 0.f, 0.f, 0.f};
            *(v4f*)lptr = z;
        }
    }

#if ASYNC_B128 || ASYNC_B32
#if __has_builtin(__builtin_amdgcn_s_wait_asynccnt)
    __builtin_amdgcn_s_wait_asynccnt(0);
#else
    asm volatile("s_wait_asynccnt 0" ::: "memory");
#endif
#endif
    __syncthreads();

    // Flipped 4x4 weights (upfirdn2d correlates with flip(kernel)): w[t] = k[15-t].
    float w[16];
#pragma unroll
    for (int t = 0; t < 16; ++t) w[t] = kern[15 - t];

    // Each thread: 1 output row x 4 consecutive output cols.
    const int tx = (tid & 15) << 2;  // 0,4,...,60
    const int ty = tid >> 4;         // 0..15

    // Output x = x0+tx+q taps input cols (x-2..x+1); LDS col of input col g is
    // g - gx_base = g - x0 + 4, so the 7-wide window starts at tx + 2.
    float a0 = 0.f, a1 = 0.f, a2 = 0.f, a3 = 0.f;
#pragma unroll
    for (int i = 0; i < 4; ++i) {
        const float* row = &tile[(ty + i) * LDS_STRIDE + tx + 2];
        const float r0 = row[0], r1 = row[1], r2 = row[2], r3 = row[3];
        const float r4 = row[4], r5 = row[5], r6 = row[6];
        const float w0 = w[i * 4 + 0], w1 = w[i * 4 + 1];
        const float w2 = w[i * 4 + 2], w3 = w[i * 4 + 3];
        a0 = fmaf(r0, w0, fmaf(r1, w1, fmaf(r2, w2, fmaf(r3, w3, a0))));
        a1 = fmaf(r1, w0, fmaf(r2, w1, fmaf(r3, w2, fmaf(r4, w3, a1))));
        a2 = fmaf(r2, w0, fmaf(r3, w1, fmaf(r4, w2, fmaf(r5, w3, a2))));
        a3 = fmaf(r3, w0, fmaf(r4, w1, fmaf(r5, w2, fmaf(r6, w3, a3))));
    }

    // Non-temporal 128-bit store: output (268MB) is write-once, keep it out of L2.
    v4f res = {a0, a1, a2, a3};
    v4f* dst = (v4f*)(oplane + (long)(y0 + ty) * W + (x0 + tx));
    __builtin_nontemporal_store(res, dst);
}

extern "C" void kernel_launch(void* const* d_in, const int* in_sizes, int n_in,
                              void* d_out, int out_size, void* d_ws, size_t ws_size,
                              hipStream_t stream) {
    const float* x    = (const float*)d_in[0];   // (8,128,256,256) fp32
    const float* kern = (const float*)d_in[1];   // (4,4) fp32
    float*       out  = (float*)d_out;

    const int H = 256, W = 256;
    const int planes = in_sizes[0] / (H * W);    // N*C = 1024

    dim3 grid(W / TILE_W, H / TILE_H, planes);   // (4, 16, 1024)
    blur4x4_kernel<<<grid, NTHREADS, 0, stream>>>(x, kern, out, H, W);
}